// GeneratorQuantumCircuit_6107443495449
// MI455X (gfx1250) — compile-verified
//
#include <hip/hip_runtime.h>
#include <hip/hip_bf16.h>
#include <math.h>

typedef __attribute__((ext_vector_type(2))) float v2f;
typedef __attribute__((ext_vector_type(4))) float v4f;
typedef __attribute__((ext_vector_type(8))) float v8f;

#define N_QUBITS 4
#define N_LAYERS 6

// ---------------------------------------------------------------------------
// Kernel 1: build the folded circuit unitary U' = U * diag(phase_k) where
// U is the 6-layer (CNOT-ring + RY) unitary and phase_k = (-i)^popcount(k).
// Thread k (k<16) evolves basis column e_k. Output: d_ws[0..255] = Re(U')
// row-major, d_ws[256..511] = Im(U') row-major.
// ---------------------------------------------------------------------------
__global__ void qc_prep(const float* __restrict__ wts, float* __restrict__ Upr) {
    const int k = threadIdx.x;
    if (k >= 16) return;

    float re[16], im[16];
#pragma unroll
    for (int i = 0; i < 16; ++i) { re[i] = (i == k) ? 1.0f : 0.0f; im[i] = 0.0f; }

#pragma unroll
    for (int l = 0; l < N_LAYERS; ++l) {
        // CNOT ring: control w, target (w+1)%4.  wire w <-> bit (3-w)
#pragma unroll
        for (int wq = 0; wq < N_QUBITS; ++wq) {
            const int cmask = 1 << (3 - wq);
            const int tmask = 1 << (3 - ((wq + 1) & 3));
            float nr[16], ni[16];
#pragma unroll
            for (int i = 0; i < 16; ++i) {
                const int src = (i & cmask) ? (i ^ tmask) : i;
                nr[i] = re[src]; ni[i] = im[src];
            }
#pragma unroll
            for (int i = 0; i < 16; ++i) { re[i] = nr[i]; im[i] = ni[i]; }
        }
        // RY(weights[l][w]) on each wire (real 2x2 rotation on amplitude pairs)
#pragma unroll
        for (int wq = 0; wq < N_QUBITS; ++wq) {
            const float h = 0.5f * wts[l * N_QUBITS + wq];
            const float c = __cosf(h), s = __sinf(h);   // native trig: |h|~0.05
            const int mask = 1 << (3 - wq);
#pragma unroll
            for (int i = 0; i < 16; ++i) {
                if (i & mask) continue;
                const int j = i | mask;
                const float ar = re[i], ai = im[i], br = re[j], bi = im[j];
                re[i] = c * ar - s * br;  im[i] = c * ai - s * bi;
                re[j] = s * ar + c * br;  im[j] = s * ai + c * bi;
            }
        }
    }

    // fold column phase (-i)^popcount(k): 0->(1,0) 1->(0,-1) 2->(-1,0) 3->(0,1)
    const int pc = __popc(k) & 3;
    const float phr = (pc == 0) ? 1.0f : (pc == 2) ? -1.0f : 0.0f;
    const float phi = (pc == 1) ? -1.0f : (pc == 3) ? 1.0f : 0.0f;
#pragma unroll
    for (int i = 0; i < 16; ++i) {
        Upr[i * 16 + k]       = re[i] * phr - im[i] * phi;   // Re(U')
        Upr[256 + i * 16 + k] = re[i] * phi + im[i] * phr;   // Im(U')
    }
}

// ---------------------------------------------------------------------------
// Kernel 2: batched 16x16 real matvec via chained v_wmma_f32_16x16x4_f32.
// Each wave handles tiles of 16 samples; both lane-halves mirror the same
// sample (n = lane%16).  B fragments are built directly from the rank-1
// butterfly structure p[4kk+j] = a4[kk]*b4[j]; the only half-dependent
// term is one select g2 = hi ? s2 : c2.
// ---------------------------------------------------------------------------
__global__ void __launch_bounds__(256)
qc_main(const float* __restrict__ x, const float* __restrict__ Upr,
        float* __restrict__ out, int nTiles) {
    const int lane = threadIdx.x & 31;
    const int wib  = threadIdx.x >> 5;
    const int gwave  = blockIdx.x * (blockDim.x >> 5) + wib;
    const int nWaves = gridDim.x * (blockDim.x >> 5);
    const bool hi = lane >= 16;           // upper half of wave
    const int  n  = lane & 15;            // sample-in-tile == A-matrix row M

    // Loop-invariant A fragments: lane supplies U'[n][4kk+2h] , U'[n][4kk+1+2h]
    v2f Ar[4], Ai[4];
    const float* UrRow = Upr + n * 16;
    const float* UiRow = Upr + 256 + n * 16;
#pragma unroll
    for (int kk = 0; kk < 4; ++kk) {
        const int c = 4 * kk + (hi ? 2 : 0);
        Ar[kk] = *(const v2f*)(UrRow + c);
        Ai[kk] = *(const v2f*)(UiRow + c);
    }

    for (int t = gwave; t < nTiles; t += nWaves) {
        const int s = t * 16 + n;                       // this lane's sample
        const v4f xv = *(const v4f*)(x + s * 4);

        // per-wire cos/sin of x/2 (native hardware trig: x is standard normal)
        const float c0 = __cosf(xv.x * 0.5f), s0 = __sinf(xv.x * 0.5f);
        const float c1 = __cosf(xv.y * 0.5f), s1 = __sinf(xv.y * 0.5f);
        const float c2 = __cosf(xv.z * 0.5f), s2 = __sinf(xv.z * 0.5f);
        const float c3 = __cosf(xv.w * 0.5f), s3 = __sinf(xv.w * 0.5f);

        // rank-1 butterfly: p[4kk+j] = a4[kk] * b4[j];  this half only needs
        // b4[2h] = (hi?s2:c2)*c3 and b4[2h+1] = (hi?s2:c2)*s3.
        const float g2 = hi ? s2 : c2;       // single cndmask
        const float bx = g2 * c3;
        const float by = g2 * s3;
        const float a40 = c0 * c1, a41 = c0 * s1, a42 = s0 * c1, a43 = s0 * s1;

        // final_real = Re(U')*p ; final_imag = Im(U')*p  (4 chained K=4 WMMAs)
        v8f accR = {}; v8f accI = {};
        {
            v2f b; b.x = a40 * bx; b.y = a40 * by;
            accR = __builtin_amdgcn_wmma_f32_16x16x4_f32(false, Ar[0], false, b, (short)0, accR, false, false);
            accI = __builtin_amdgcn_wmma_f32_16x16x4_f32(false, Ai[0], false, b, (short)0, accI, false, false);
        }
        {
            v2f b; b.x = a41 * bx; b.y = a41 * by;
            accR = __builtin_amdgcn_wmma_f32_16x16x4_f32(false, Ar[1], false, b, (short)0, accR, false, false);
            accI = __builtin_amdgcn_wmma_f32_16x16x4_f32(false, Ai[1], false, b, (short)0, accI, false, false);
        }
        {
            v2f b; b.x = a42 * bx; b.y = a42 * by;
            accR = __builtin_amdgcn_wmma_f32_16x16x4_f32(false, Ar[2], false, b, (short)0, accR, false, false);
            accI = __builtin_amdgcn_wmma_f32_16x16x4_f32(false, Ai[2], false, b, (short)0, accI, false, false);
        }
        {
            v2f b; b.x = a43 * bx; b.y = a43 * by;
            accR = __builtin_amdgcn_wmma_f32_16x16x4_f32(false, Ar[3], false, b, (short)0, accR, false, false);
            accI = __builtin_amdgcn_wmma_f32_16x16x4_f32(false, Ai[3], false, b, (short)0, accI, false, false);
        }

        // probabilities for rows j = v + 8h of sample n
        float pr[8];
#pragma unroll
        for (int v = 0; v < 8; ++v)
            pr[v] = accR[v] * accR[v] + accI[v] * accI[v];

        // partial Walsh butterfly: 4 signed sums in 18 adds, depth 3.
        //   sum = S pr[v]; t1 sign=bit2(v); t2 sign=bit1(v); t3 sign=bit0(v)
        const float s01 = pr[0] + pr[1], d01 = pr[0] - pr[1];
        const float s23 = pr[2] + pr[3], d23 = pr[2] - pr[3];
        const float s45 = pr[4] + pr[5], d45 = pr[4] - pr[5];
        const float s67 = pr[6] + pr[7], d67 = pr[6] - pr[7];
        const float e03 = s01 + s23, m03 = s01 - s23, q03 = d01 + d23;
        const float e47 = s45 + s67, m47 = s45 - s67, q47 = d45 + d67;
        const float sum = e03 + e47;
        const float t1  = e03 - e47;
        const float t2  = m03 + m47;
        const float t3  = q03 + q47;

        float r0 = hi ? -sum : sum;          // w=0 sign = bit3(j) = h
        // combine the two lane-halves (rows 0..7 with rows 8..15)
        r0 += __shfl_xor(r0, 16, 32);
        float r1 = t1 + __shfl_xor(t1, 16, 32);
        float r2 = t2 + __shfl_xor(t2, 16, 32);
        float r3 = t3 + __shfl_xor(t3, 16, 32);

        if (!hi) {
            v4f o; o.x = r0; o.y = r1; o.z = r2; o.w = r3;
            *(v4f*)(out + s * 4) = o;
        }
    }
}

// ---------------------------------------------------------------------------
extern "C" void kernel_launch(void* const* d_in, const int* in_sizes, int n_in,
                              void* d_out, int out_size, void* d_ws, size_t ws_size,
                              hipStream_t stream) {
    const float* x   = (const float*)d_in[0];   // (B,4) f32
    const float* wts = (const float*)d_in[1];   // (6,4) f32
    float* out = (float*)d_out;                 // (B,4) f32
    float* Upr = (float*)d_ws;                  // 512 f32: Re(U'), Im(U')

    const int B = in_sizes[0] / N_QUBITS;
    const int nTiles = B / 16;

    qc_prep<<<1, 32, 0, stream>>>(wts, Upr);

    int blocks = 1024;
    if (blocks * 8 > nTiles) blocks = (nTiles + 7) / 8 > 0 ? (nTiles + 7) / 8 : 1;
    qc_main<<<blocks, 256, 0, stream>>>(x, Upr, out, nTiles);
}